// DeformConv_10986526343480
// MI455X (gfx1250) — compile-verified
//
#include <hip/hip_runtime.h>
#include <hip/hip_bf16.h>

typedef __attribute__((ext_vector_type(2))) float v2f;
typedef __attribute__((ext_vector_type(8))) float v8f;

#define BATCH 4
#define CIN   64
#define COUT  64
#define HH    256
#define WW    256
#define HP    258            // padded
#define NPTS  9
#define K_TOT 576            // NPTS * CIN, k = n*64 + c
#define ASTRIDE 580          // 576 + 4 pad -> conflict-free LDS columns

// ---------------------------------------------------------------------------
// 1) pack x (b,c,h,w) -> zero-padded NHWC xn (b,258,258,64)
// ---------------------------------------------------------------------------
__global__ void pack_nhwc(const float* __restrict__ x, float* __restrict__ xn) {
    int t = blockIdx.x * blockDim.x + threadIdx.x;
    const int total = BATCH * HP * HP;
    if (t >= total) return;
    int wp = t % HP; int tmp = t / HP; int hp = tmp % HP; int b = tmp / HP;
    float* dst = xn + (size_t)t * CIN;
    if (hp >= 1 && hp <= HH && wp >= 1 && wp <= WW) {
        const float* src = x + (size_t)b * CIN * HH * WW + (size_t)(hp - 1) * WW + (wp - 1);
        #pragma unroll 8
        for (int c = 0; c < CIN; ++c) dst[c] = src[(size_t)c * (HH * WW)];
    } else {
        float4 z = make_float4(0.f, 0.f, 0.f, 0.f);
        float4* d4 = (float4*)dst;
        #pragma unroll
        for (int i = 0; i < CIN / 4; ++i) d4[i] = z;
    }
}

// ---------------------------------------------------------------------------
// 2a) reorder main conv weights (o,c,n) -> paired B layout:
//     brs[((k>>1)*64 + o)*2 + (k&1)] with k = n*64 + c
// ---------------------------------------------------------------------------
__global__ void reorder_B(const float* __restrict__ cw, float* __restrict__ brs) {
    int idx = blockIdx.x * blockDim.x + threadIdx.x;      // 36864
    int o = idx & 63;
    int k = idx >> 6;
    int n = k >> 6;          // k = n*64 + c
    int c = k & 63;
    float v = cw[(size_t)o * K_TOT + c * NPTS + n];
    brs[(((size_t)(k >> 1) * 64 + o) << 1) + (k & 1)] = v;
}

// 2b) reorder offset-conv weights (18,64,3,3) -> wr[tap*64*18 + c*18 + oc]
__global__ void reorder_W(const float* __restrict__ pw, float* __restrict__ wr) {
    int idx = blockIdx.x * blockDim.x + threadIdx.x;      // 10368
    if (idx >= 18 * CIN * NPTS) return;
    int oc  = idx % 18;
    int c   = (idx / 18) % CIN;
    int tap = idx / (18 * CIN);
    wr[idx] = pw[(size_t)oc * (CIN * NPTS) + c * NPTS + tap];
}

// ---------------------------------------------------------------------------
// 3) offset conv: 18 channels per pixel, reads padded NHWC (float4), weights
//    in LDS; writes off[(b*65536+h*256+w)*18 + oc]
// ---------------------------------------------------------------------------
__global__ void __launch_bounds__(256)
offset_conv(const float* __restrict__ xn, const float* __restrict__ wr_g,
            const float* __restrict__ bias, float* __restrict__ off) {
    __shared__ float wr[NPTS * CIN * 18];                 // 41472 B
    for (int i = threadIdx.x; i < NPTS * CIN * 18; i += 256) wr[i] = wr_g[i];
    __syncthreads();

    int m = blockIdx.x * 256 + threadIdx.x;               // 0..262143
    int w = m & 255, h = (m >> 8) & 255, b = m >> 16;

    float acc[18];
    #pragma unroll
    for (int o = 0; o < 18; ++o) acc[o] = bias[o];

    #pragma unroll
    for (int ky = 0; ky < 3; ++ky)
        #pragma unroll
        for (int kx = 0; kx < 3; ++kx) {
            const float4* xp =
                (const float4*)(xn + (((size_t)b * HP + (h + ky)) * HP + (w + kx)) * CIN);
            const float* wp = wr + (ky * 3 + kx) * (CIN * 18);
            for (int c4 = 0; c4 < CIN / 4; ++c4) {
                float4 xv = xp[c4];
                const float* w0 = wp + (c4 * 4) * 18;
                #pragma unroll
                for (int o = 0; o < 18; ++o) {
                    acc[o] += xv.x * w0[o] + xv.y * w0[18 + o]
                            + xv.z * w0[36 + o] + xv.w * w0[54 + o];
                }
            }
        }
    float* dst = off + (size_t)m * 18;
    #pragma unroll
    for (int o = 0; o < 18; ++o) dst[o] = acc[o];
}

// ---------------------------------------------------------------------------
// 4) fused bilinear-sample + WMMA GEMM.  Block = 128 thr (4 waves) = 16 pixels.
//    Wave w computes D[16 px x 16 outch], out channels 16w..16w+15.
// ---------------------------------------------------------------------------
__global__ void __launch_bounds__(128)
deform_gemm(const float* __restrict__ xn, const float* __restrict__ off,
            const float* __restrict__ brs, float* __restrict__ out) {
    __shared__ float Atile[16 * ASTRIDE];                 // 37120 B
    __shared__ int   mAddr[144 * 4];
    __shared__ float mWgt[144 * 4];

    const int tid = threadIdx.x;
    const int m_base = blockIdx.x * 16;

    // ---- phase A: sampling descriptors for 16 px x 9 taps ----
    for (int idx = tid; idx < 144; idx += 128) {
        int ml = idx / 9;
        int n  = idx - ml * 9;
        int gm = m_base + ml;
        int w = gm & 255, h = (gm >> 8) & 255, b = gm >> 16;
        float ox = off[(size_t)gm * 18 + n];
        float oy = off[(size_t)gm * 18 + 9 + n];
        float p_x = ox + (float)(h + 1) + ((float)(n / 3) - 1.0f);
        float p_y = oy + (float)(w + 1) + ((float)(n % 3) - 1.0f);
        float fx = floorf(p_x), fy = floorf(p_y);
        float ltxf = fminf(fmaxf(fx,        0.f), 257.f);
        float ltyf = fminf(fmaxf(fy,        0.f), 257.f);
        float rbxf = fminf(fmaxf(fx + 1.f,  0.f), 257.f);
        float rbyf = fminf(fmaxf(fy + 1.f,  0.f), 257.f);
        float px   = fminf(fmaxf(p_x,       0.f), 257.f);
        float py   = fminf(fmaxf(p_y,       0.f), 257.f);
        int ltx = (int)ltxf, lty = (int)ltyf, rbx = (int)rbxf, rby = (int)rbyf;
        float wx0 = 1.f + (ltxf - px);
        float wx1 = 1.f - (rbxf - px);
        float wy0 = 1.f + (ltyf - py);
        float wy1 = 1.f - (rbyf - py);
        int bb = b * HP;
        mAddr[idx * 4 + 0] = ((bb + ltx) * HP + lty) * CIN;
        mAddr[idx * 4 + 1] = ((bb + rbx) * HP + rby) * CIN;
        mAddr[idx * 4 + 2] = ((bb + ltx) * HP + rby) * CIN;
        mAddr[idx * 4 + 3] = ((bb + rbx) * HP + lty) * CIN;
        mWgt[idx * 4 + 0] = wx0 * wy0;
        mWgt[idx * 4 + 1] = wx1 * wy1;
        mWgt[idx * 4 + 2] = wx0 * wy1;
        mWgt[idx * 4 + 3] = wx1 * wy0;
    }
    __syncthreads();

    // ---- phase B: build 16 x 576 f32 A tile; float4 over channels ----
    // task e covers 4 channels: c4 = e & 15, mn = e >> 4 (= ml*9 + n)
    {
        const float4* x4 = (const float4*)xn;
        for (int e = tid; e < 16 * K_TOT / 4; e += 128) {
            int c4 = e & 15;
            int mn = e >> 4;
            int ml = mn / 9;
            int n  = mn - ml * 9;
            int a0 = mAddr[mn * 4 + 0] >> 2, a1 = mAddr[mn * 4 + 1] >> 2;
            int a2 = mAddr[mn * 4 + 2] >> 2, a3 = mAddr[mn * 4 + 3] >> 2;
            float w0 = mWgt[mn * 4 + 0], w1 = mWgt[mn * 4 + 1];
            float w2 = mWgt[mn * 4 + 2], w3 = mWgt[mn * 4 + 3];
            float4 p0 = x4[a0 + c4], p1 = x4[a1 + c4];
            float4 p2 = x4[a2 + c4], p3 = x4[a3 + c4];
            float4 v;
            v.x = w0 * p0.x + w1 * p1.x + w2 * p2.x + w3 * p3.x;
            v.y = w0 * p0.y + w1 * p1.y + w2 * p2.y + w3 * p3.y;
            v.z = w0 * p0.z + w1 * p1.z + w2 * p2.z + w3 * p3.z;
            v.w = w0 * p0.w + w1 * p1.w + w2 * p2.w + w3 * p3.w;
            float4* At4 = (float4*)&Atile[ml * ASTRIDE + n * 64];
            At4[c4] = v;
        }
    }
    __syncthreads();

    // ---- phase C: 144 steps of V_WMMA_F32_16X16X4_F32 per wave ----
    const int lane  = tid & 31;
    const int wave  = tid >> 5;
    const int row   = lane & 15;
    const int hi    = lane >> 4;             // 0: K0/K1, 1: K2/K3
    const int o_idx = wave * 16 + row;

    v8f acc = {};
    const float* aBase = &Atile[row * ASTRIDE + 2 * hi];
    const v2f*   bB    = (const v2f*)brs;

    #pragma unroll 4
    for (int kt = 0; kt < 144; ++kt) {
        int k0 = kt * 4;
        v2f a = *(const v2f*)(aBase + k0);                     // ds_load_b64
        v2f b = bB[(size_t)((k0 >> 1) + hi) * 64 + o_idx];     // global_load_b64
        acc = __builtin_amdgcn_wmma_f32_16x16x4_f32(
                  false, a, false, b, (short)0, acc, false, false);
    }

    // ---- store D in bohw order: lane rows are 8 consecutive hw ----
    {
        int b  = m_base >> 16;
        int hw = (m_base & 65535) + hi * 8;
        float* dst = out + ((size_t)(b * COUT + o_idx)) * (HH * WW) + hw;
        #pragma unroll
        for (int v = 0; v < 8; ++v) dst[v] = acc[v];
    }
}

// ---------------------------------------------------------------------------
// 5) deterministic per-plane partial stats — 256 blocks, one (b,o) plane each
// ---------------------------------------------------------------------------
__global__ void __launch_bounds__(256)
bn_stats(const float* __restrict__ out, float* __restrict__ psum, float* __restrict__ psq) {
    __shared__ float ssum[256], ssq[256];
    int bo = blockIdx.x;                                  // = b*64 + o
    float s1 = 0.f, s2 = 0.f;
    const float4* p = (const float4*)(out + (size_t)bo * (HH * WW));
    for (int i = threadIdx.x; i < (HH * WW) / 4; i += 256) {
        float4 v = p[i];
        s1 += v.x + v.y + v.z + v.w;
        s2 += v.x * v.x + v.y * v.y + v.z * v.z + v.w * v.w;
    }
    ssum[threadIdx.x] = s1; ssq[threadIdx.x] = s2;
    __syncthreads();
    for (int s = 128; s > 0; s >>= 1) {
        if (threadIdx.x < s) {
            ssum[threadIdx.x] += ssum[threadIdx.x + s];
            ssq[threadIdx.x]  += ssq[threadIdx.x + s];
        }
        __syncthreads();
    }
    if (threadIdx.x == 0) { psum[bo] = ssum[0]; psq[bo] = ssq[0]; }
}

// ---------------------------------------------------------------------------
// 6) BN affine + LeakyReLU(0.1), in place on d_out (float4); combines the
//    4 per-batch partials deterministically
// ---------------------------------------------------------------------------
__global__ void __launch_bounds__(256)
bn_finalize(float* __restrict__ out, const float* __restrict__ psum,
            const float* __restrict__ psq, const float* __restrict__ gamma,
            const float* __restrict__ beta) {
    int i4 = blockIdx.x * 256 + threadIdx.x;      // float4 index, plane = 16384
    int o  = (i4 >> 14) & 63;
    float s1 = psum[o] + psum[64 + o] + psum[128 + o] + psum[192 + o];
    float s2 = psq[o]  + psq[64 + o]  + psq[128 + o]  + psq[192 + o];
    const float invM = 1.0f / (float)(BATCH * HH * WW);
    float mean = s1 * invM;
    float var  = s2 * invM - mean * mean;
    float sc = gamma[o] * rsqrtf(var + 1e-5f);
    float sh = beta[o] - mean * sc;
    float4* p = (float4*)out + i4;
    float4 v = *p;
    float y0 = v.x * sc + sh, y1 = v.y * sc + sh, y2 = v.z * sc + sh, y3 = v.w * sc + sh;
    v.x = y0 > 0.f ? y0 : 0.1f * y0;
    v.y = y1 > 0.f ? y1 : 0.1f * y1;
    v.z = y2 > 0.f ? y2 : 0.1f * y2;
    v.w = y3 > 0.f ? y3 : 0.1f * y3;
    *p = v;
}

// ---------------------------------------------------------------------------
extern "C" void kernel_launch(void* const* d_in, const int* in_sizes, int n_in,
                              void* d_out, int out_size, void* d_ws, size_t ws_size,
                              hipStream_t stream) {
    const float* x     = (const float*)d_in[0];   // (4,64,256,256)
    const float* pcw   = (const float*)d_in[1];   // (18,64,3,3)
    const float* pcb   = (const float*)d_in[2];   // (18,)
    const float* cw    = (const float*)d_in[3];   // (64,64,3,3)
    const float* gamma = (const float*)d_in[4];   // (64,)
    const float* beta  = (const float*)d_in[5];   // (64,)
    float* out = (float*)d_out;                   // (4,64,256,256)

    // workspace layout (floats)
    const size_t XN_N  = (size_t)BATCH * HP * HP * CIN;       // 17,040,384
    const size_t OFF_N = (size_t)BATCH * HH * WW * 18;        //  4,718,592
    const size_t BRS_N = (size_t)K_TOT * COUT;                //     36,864
    const size_t WR_N  = (size_t)NPTS * CIN * 18;             //     10,368
    float* xn   = (float*)d_ws;
    float* off  = xn  + XN_N;
    float* brs  = off + OFF_N;
    float* wr   = brs + BRS_N;
    float* psum = wr  + WR_N;                                 // 256 partials
    float* psq  = psum + 256;

    // weight reorders
    reorder_B<<<(int)(BRS_N / 256), 256, 0, stream>>>(cw, brs);
    reorder_W<<<(int)((WR_N + 255) / 256), 256, 0, stream>>>(pcw, wr);

    // pack padded NHWC
    {
        int total = BATCH * HP * HP;
        pack_nhwc<<<(total + 255) / 256, 256, 0, stream>>>(x, xn);
    }

    // offset conv (262144 px)
    offset_conv<<<1024, 256, 0, stream>>>(xn, wr, pcb, off);

    // fused sample + WMMA GEMM (16384 tiles of 16 px)
    deform_gemm<<<16384, 128, 0, stream>>>(xn, off, brs, out);

    // batch-norm statistics + finalize
    bn_stats<<<256, 256, 0, stream>>>(out, psum, psq);
    bn_finalize<<<(out_size / 4 + 255) / 256, 256, 0, stream>>>(out, psum, psq, gamma, beta);
}